// Low_Rank_RNN_54314156425434
// MI455X (gfx1250) — compile-verified
//
#include <hip/hip_runtime.h>
#include <cstddef>

// Problem dims (fixed by the reference)
#define BQ 256
#define SQ 512
#define IQ 3
#define HQ 512
#define OQ 3

typedef float v2f __attribute__((ext_vector_type(2)));
typedef float v8f __attribute__((ext_vector_type(8)));

__device__ __forceinline__ float fast_tanh(float t) {
#if __has_builtin(__builtin_amdgcn_tanhf)
  return __builtin_amdgcn_tanhf(t);   // V_TANH_F32 (gfx1250 trans pipe)
#else
  return tanhf(t);
#endif
}

// ---------------------------------------------------------------------------
// Kernel 1: fused input-projection + low-rank scan.
// One wave32 per batch element; lane owns 16 contiguous hidden units in VGPRs.
// Scalar FMAs (VOPD dual-issue pairs them).
// Writes r_out [B,S,H] and h_last [B,H].
// ---------------------------------------------------------------------------
__global__ __launch_bounds__(32) void lr_rnn_scan(
    const float* __restrict__ x,   // [B,S,I]
    const float* __restrict__ U,   // [R,H]
    const float* __restrict__ V,   // [R,H]
    const float* __restrict__ Wi,  // [H,I]
    const float* __restrict__ bi,  // [H]
    float* __restrict__ r_out,     // [B,S,H]
    float* __restrict__ h_last)    // [B,H]
{
  const int b    = blockIdx.x;     // 0..255
  const int lane = threadIdx.x;    // 0..31
  const int h0   = lane * 16;      // this lane's hidden slice

  float wi0[16], wi1[16], wi2[16];
  float uu0[16], uu1[16], vv0[16], vv1[16], bv[16], h[16];
#pragma unroll
  for (int j = 0; j < 16; ++j) {
    const int hh = h0 + j;
    wi0[j] = Wi[hh * 3 + 0];
    wi1[j] = Wi[hh * 3 + 1];
    wi2[j] = Wi[hh * 3 + 2];
    uu0[j] = U[hh];
    uu1[j] = U[HQ + hh];
    vv0[j] = V[hh];
    vv1[j] = V[HQ + hh];
    bv[j]  = bi[hh];
    h[j]   = 0.0f;
  }

  const float* xp = x + (size_t)b * SQ * IQ;
  float*       rp = r_out + (size_t)b * SQ * HQ + h0;

  float p0 = 0.0f, p1 = 0.0f;                 // h0 = 0 -> projections start at 0
  float x0 = xp[0], x1 = xp[1], x2 = xp[2];   // step-0 input, pipelined below

  for (int s = 0; s < SQ; ++s) {
    // prefetch next step's input (clamped index -> branch-free, uniform loads)
    const int   sn = (s + 1 < SQ) ? (s + 1) : (SQ - 1);
    const float* q = xp + (size_t)sn * IQ;
    const float nx0 = q[0], nx1 = q[1], nx2 = q[2];

    float q0a = 0.f, q0b = 0.f, q1a = 0.f, q1b = 0.f;
#pragma unroll
    for (int j = 0; j < 16; ++j) {
      // u = Wi·x + bi, plus rank-2 recurrence p0*U0 + p1*U1
      float t = __builtin_fmaf(wi0[j], x0, bv[j]);
      t = __builtin_fmaf(wi1[j], x1, t);
      t = __builtin_fmaf(wi2[j], x2, t);
      t = __builtin_fmaf(p0, uu0[j], t);
      t = __builtin_fmaf(p1, uu1[j], t);
      const float hn = fast_tanh(t);
      h[j] = hn;
      // partial dot products with V rows (two accumulators each to cut chain depth)
      if (j & 1) {
        q0b = __builtin_fmaf(hn, vv0[j], q0b);
        q1b = __builtin_fmaf(hn, vv1[j], q1b);
      } else {
        q0a = __builtin_fmaf(hn, vv0[j], q0a);
        q1a = __builtin_fmaf(hn, vv1[j], q1a);
      }
    }

    // stream r_out[b, s, h0:h0+16] (64B-aligned, off the critical path)
    float4* o4 = (float4*)(rp + (size_t)s * HQ);
    o4[0] = make_float4(h[0],  h[1],  h[2],  h[3]);
    o4[1] = make_float4(h[4],  h[5],  h[6],  h[7]);
    o4[2] = make_float4(h[8],  h[9],  h[10], h[11]);
    o4[3] = make_float4(h[12], h[13], h[14], h[15]);

    // cross-lane reduction: p = h · V^T (rank-2), result in all lanes
    float q0 = q0a + q0b;
    float q1 = q1a + q1b;
#pragma unroll
    for (int off = 16; off > 0; off >>= 1) {
      q0 += __shfl_xor(q0, off, 32);
      q1 += __shfl_xor(q1, off, 32);
    }
    p0 = q0; p1 = q1;
    x0 = nx0; x1 = nx1; x2 = nx2;
  }

  float4* hl = (float4*)(h_last + (size_t)b * HQ + h0);
  hl[0] = make_float4(h[0],  h[1],  h[2],  h[3]);
  hl[1] = make_float4(h[4],  h[5],  h[6],  h[7]);
  hl[2] = make_float4(h[8],  h[9],  h[10], h[11]);
  hl[3] = make_float4(h[12], h[13], h[14], h[15]);
}

// ---------------------------------------------------------------------------
// Kernel 2: output projection out = r_out @ Wo^T + bo via V_WMMA_F32_16X16X4_F32.
// One wave per 32-row tile (two 16x16 accumulators sharing each A fragment).
// A = zero-padded Wo staged in LDS (no EXEC divergence in the hot loop);
// B fragments batched 8-deep before the WMMA chain to hide load latency.
// ---------------------------------------------------------------------------
__global__ __launch_bounds__(256) void lr_rnn_outproj(
    const float* __restrict__ r,   // [B*S, H]
    const float* __restrict__ Wo,  // [O, H]
    const float* __restrict__ bo,  // [O]
    float* __restrict__ out)       // [B*S, O]
{
  __shared__ float woPad[16 * HQ];          // 32 KB: rows O..15 zero-padded

  for (int i = threadIdx.x; i < 16 * HQ; i += 256) {
    const int m = i / HQ;
    const int k = i - m * HQ;
    woPad[i] = (m < OQ) ? Wo[m * HQ + k] : 0.0f;
  }
  __syncthreads();

  const int lane   = threadIdx.x & 31;
  const int wavein = threadIdx.x >> 5;
  const int tile   = blockIdx.x * 8 + wavein;   // 512 blocks * 8 waves = 4096
  const int row0   = tile * 32;                 // 32 rows per wave

  const int m  = lane & 15;          // A row (out channel) / B column (batch row)
  const int kh = (lane >> 4) << 1;   // lanes 0-15 carry K={0,1}; 16-31 carry K={2,3}

  // 8-byte aligned v2f views: one b64 load per fragment half
  const v2f* rb0 = (const v2f*)(r + (size_t)(row0 + m) * HQ + kh);
  const v2f* rb1 = (const v2f*)(r + (size_t)(row0 + 16 + m) * HQ + kh);
  const v2f* wb  = (const v2f*)(&woPad[m * HQ + kh]);

  v8f acc0 = {};
  v8f acc1 = {};
  for (int k0 = 0; k0 < HQ; k0 += 32) {       // 8 K-chunks of 4 per block
    v2f bf0[8], bf1[8], af[8];
    const int base = k0 >> 1;                 // v2f index
#pragma unroll
    for (int c = 0; c < 8; ++c) bf0[c] = rb0[base + 2 * c];
#pragma unroll
    for (int c = 0; c < 8; ++c) bf1[c] = rb1[base + 2 * c];
#pragma unroll
    for (int c = 0; c < 8; ++c) af[c]  = wb[base + 2 * c];
#pragma unroll
    for (int c = 0; c < 8; ++c) {
      acc0 = __builtin_amdgcn_wmma_f32_16x16x4_f32(
          false, af[c], false, bf0[c], (short)0, acc0, false, false);
      acc1 = __builtin_amdgcn_wmma_f32_16x16x4_f32(
          false, af[c], false, bf1[c], (short)0, acc1, false, false);
    }
  }

  const float b0 = bo[0], b1 = bo[1], b2 = bo[2];
  if (lane < 16) {
    const size_t rowA = (size_t)(row0 + lane);
    out[rowA * OQ + 0] = acc0[0] + b0;
    out[rowA * OQ + 1] = acc0[1] + b1;
    out[rowA * OQ + 2] = acc0[2] + b2;
    const size_t rowB = rowA + 16;
    out[rowB * OQ + 0] = acc1[0] + b0;
    out[rowB * OQ + 1] = acc1[1] + b1;
    out[rowB * OQ + 2] = acc1[2] + b2;
  }
}

// ---------------------------------------------------------------------------
// Launch
// ---------------------------------------------------------------------------
extern "C" void kernel_launch(void* const* d_in, const int* in_sizes, int n_in,
                              void* d_out, int out_size, void* d_ws, size_t ws_size,
                              hipStream_t stream) {
  (void)in_sizes; (void)n_in; (void)out_size; (void)d_ws; (void)ws_size;

  const float* x  = (const float*)d_in[0];  // [B,S,I]
  const float* U  = (const float*)d_in[1];  // [R,H]
  const float* V  = (const float*)d_in[2];  // [R,H]
  const float* Wi = (const float*)d_in[3];  // [H,I]
  const float* bi = (const float*)d_in[4];  // [H]
  const float* Wo = (const float*)d_in[5];  // [O,H]
  const float* bo = (const float*)d_in[6];  // [O]

  // d_out layout: output [B,S,O] | h_last [B,H] | r_out [B,S,H]
  float* out    = (float*)d_out;
  float* h_last = out + (size_t)BQ * SQ * OQ;
  float* r_out  = h_last + (size_t)BQ * HQ;

  lr_rnn_scan<<<BQ, 32, 0, stream>>>(x, U, V, Wi, bi, r_out, h_last);

  const int tiles  = (BQ * SQ) / 32;  // 4096 waves, 32 rows each
  const int blocks = tiles / 8;       // 8 waves per 256-thread block
  lr_rnn_outproj<<<blocks, 256, 0, stream>>>(r_out, Wo, bo, out);
}